// GAT_4784593568512
// MI455X (gfx1250) — compile-verified
//
#include <hip/hip_runtime.h>
#include <math.h>

// ---------------------------------------------------------------------------
// GATv2 x2 for gfx1250 (wave32). Constants from the reference problem.
// ---------------------------------------------------------------------------
#define FDIM   128   // input features / HC (also GEMM K)
#define HEADS  4
#define CH     32    // channels per head (layer 1)
#define EDIM   16    // edge attr dim
#define O2     4     // layer-2 H*NCLS
#define SLOPE_ATT 0.2f
#define SLOPE_ACT 0.01f

typedef __attribute__((ext_vector_type(2))) float v2f;
typedef __attribute__((ext_vector_type(8))) float v8f;

__device__ __forceinline__ float leaky(float v, float s) {
    return v > 0.f ? v : s * v;
}

// monotonic-encoding float atomic max (works for mixed signs)
__device__ __forceinline__ void atomic_max_f(float* addr, float v) {
    if (v >= 0.f) atomicMax((int*)addr, __float_as_int(v));
    else          atomicMin((unsigned int*)addr, __float_as_uint(v));
}

// ---------------------------------------------------------------------------
// generic fill
// ---------------------------------------------------------------------------
__global__ void fill_kernel(float* __restrict__ p, float v, long n) {
    long i = (long)blockIdx.x * blockDim.x + threadIdx.x;
    if (i < n) p[i] = v;
}

// ---------------------------------------------------------------------------
// deg + sum of incoming edge attrs (for self-loop fill_value='mean')
// ---------------------------------------------------------------------------
__global__ void edge_prep_kernel(const int* __restrict__ dst,
                                 const float* __restrict__ edge_attr,
                                 float* __restrict__ deg,
                                 float* __restrict__ loopA, int E) {
    int e = blockIdx.x * blockDim.x + threadIdx.x;
    if (e >= E) return;
    int d = dst[e];
    atomicAdd(&deg[d], 1.f);
    const float4* ea4 = (const float4*)(edge_attr + (long)e * EDIM);
#pragma unroll
    for (int q = 0; q < EDIM / 4; ++q) {
        float4 v = ea4[q];
        atomicAdd(&loopA[(long)d * EDIM + 4 * q + 0], v.x);
        atomicAdd(&loopA[(long)d * EDIM + 4 * q + 1], v.y);
        atomicAdd(&loopA[(long)d * EDIM + 4 * q + 2], v.z);
        atomicAdd(&loopA[(long)d * EDIM + 4 * q + 3], v.w);
    }
}

__global__ void loop_norm_kernel(float* __restrict__ loopA,
                                 const float* __restrict__ deg, int N) {
    int i = blockIdx.x * blockDim.x + threadIdx.x;
    if (i >= N) return;
    float inv = 1.f / fmaxf(deg[i], 1.f);
#pragma unroll
    for (int k = 0; k < EDIM; ++k) loopA[(long)i * EDIM + k] *= inv;
}

// ---------------------------------------------------------------------------
// fp32 GEMM via V_WMMA_F32_16X16X4_F32 : D[M x Nout] = A[M x K=128] * B[K x Nout]
// One wave per 16x16 output tile; K is compile-time so the 32 WMMA steps fully
// unroll and loads are hoisted. OOB handled by clamped addresses + value
// select (v_cndmask) -- no divergent exec-masked loads.
// A layout (16x4 f32): lane l -> row=l&15; vgpr j -> K = 2*(l>>4)+j
// B layout (4x16 f32): lane l -> col=l&15; vgpr j -> K = 2*(l>>4)+j
// C/D layout        : lane l -> col=l&15; vgpr j -> row = j + 8*(l>>4)
// ---------------------------------------------------------------------------
__global__ void gemm_f32_wmma(const float* __restrict__ A,
                              const float* __restrict__ B,
                              float* __restrict__ D,
                              int M, int Nout) {
    const int K = FDIM;
    int ntiles = (Nout + 15) >> 4;
    int mtiles = (M + 15) >> 4;
    int tile = blockIdx.x * (blockDim.x >> 5) + (threadIdx.x >> 5);
    if (tile >= mtiles * ntiles) return;
    int tm = tile / ntiles, tn = tile - tm * ntiles;
    int lane = threadIdx.x & 31;
    int half = lane >> 4;
    int r    = lane & 15;
    int arow = tm * 16 + r;
    int bcol = tn * 16 + r;
    bool arow_ok = arow < M;
    bool bcol_ok = bcol < Nout;
    // clamped base pointers: loads always in-bounds, values zeroed by select
    const float* Arow = A + (long)(arow_ok ? arow : 0) * K + 2 * half;
    const float* Bcol = B + (bcol_ok ? bcol : 0) + (long)(2 * half) * Nout;

    v8f acc = {};
#pragma unroll
    for (int k0 = 0; k0 < K; k0 += 4) {
        v2f a = *(const v2f*)(Arow + k0);          // global_load_b64
        v2f b;
        b.x = Bcol[(long)k0 * Nout];
        b.y = Bcol[(long)(k0 + 1) * Nout];
        a.x = arow_ok ? a.x : 0.f;                 // v_cndmask, no branch
        a.y = arow_ok ? a.y : 0.f;
        b.x = bcol_ok ? b.x : 0.f;
        b.y = bcol_ok ? b.y : 0.f;
        acc = __builtin_amdgcn_wmma_f32_16x16x4_f32(
            false, a, false, b, (short)0, acc, false, false);
    }
    if (bcol_ok) {
#pragma unroll
        for (int j = 0; j < 8; ++j) {
            int m = tm * 16 + j + 8 * half;
            if (m < M) D[(long)m * Nout + bcol] = acc[j];
        }
    }
}

// ---------------------------------------------------------------------------
// Layer-1 edge pass 1: logits + per-dst atomic max. One wave per edge.
// lane = channel-within-head; wave reduces over C=32 with shfl_xor.
// ---------------------------------------------------------------------------
__global__ void edge_logits1_kernel(const int* __restrict__ src,
                                    const int* __restrict__ dst,
                                    const float* __restrict__ edge_attr,
                                    const float* __restrict__ loopA,
                                    const float* __restrict__ xl,
                                    const float* __restrict__ xr,
                                    const float* __restrict__ We,
                                    const float* __restrict__ att,
                                    float* __restrict__ logits,
                                    float* __restrict__ lmax,
                                    int E, int N) {
    __shared__ float sWe[EDIM * FDIM];   // 8 KB
    __shared__ float sAtt[FDIM];
    for (int i = threadIdx.x; i < EDIM * FDIM; i += blockDim.x) sWe[i] = We[i];
    for (int i = threadIdx.x; i < FDIM; i += blockDim.x) sAtt[i] = att[i];
    __syncthreads();

    int EN = E + N;
    int lane = threadIdx.x & 31;
    int eidx = blockIdx.x * (blockDim.x >> 5) + (threadIdx.x >> 5);
    if (eidx >= EN) return;

    int s, d;
    const float* ea;
    if (eidx < E) { s = src[eidx]; d = dst[eidx]; ea = edge_attr + (long)eidx * EDIM; }
    else          { s = d = eidx - E;             ea = loopA + (long)s * EDIM; }

    float ear[EDIM];
    const float4* ea4 = (const float4*)ea;        // rows are 64B aligned
#pragma unroll
    for (int q = 0; q < EDIM / 4; ++q) {          // wave-uniform -> broadcast b128
        float4 v = ea4[q];
        ear[4 * q + 0] = v.x; ear[4 * q + 1] = v.y;
        ear[4 * q + 2] = v.z; ear[4 * q + 3] = v.w;
    }

#pragma unroll
    for (int h = 0; h < HEADS; ++h) {
        int ch = h * CH + lane;
        float v = xl[(long)s * FDIM + ch] + xr[(long)d * FDIM + ch];
        float ew = 0.f;
#pragma unroll
        for (int k = 0; k < EDIM; ++k) ew = fmaf(ear[k], sWe[k * FDIM + ch], ew);
        v = leaky(v + ew, SLOPE_ATT);
        float p = v * sAtt[ch];
#pragma unroll
        for (int off = 16; off; off >>= 1) p += __shfl_xor(p, off, 32);
        if (lane == 0) {
            logits[(long)eidx * HEADS + h] = p;
            atomic_max_f(&lmax[(long)d * HEADS + h], p);
        }
    }
}

// ---------------------------------------------------------------------------
// Shared (both layers): denom accumulation. One thread per edge.
// ---------------------------------------------------------------------------
__global__ void edge_denom_kernel(const float* __restrict__ logits,
                                  const float* __restrict__ lmax,
                                  float* __restrict__ denom,
                                  const int* __restrict__ dst, int E, int N) {
    int EN = E + N;
    int idx = blockIdx.x * blockDim.x + threadIdx.x;
    if (idx >= EN) return;
    int d = (idx < E) ? dst[idx] : (idx - E);
#pragma unroll
    for (int h = 0; h < HEADS; ++h) {
        float ex = expf(logits[(long)idx * HEADS + h] - lmax[(long)d * HEADS + h]);
        atomicAdd(&denom[(long)d * HEADS + h], ex);
    }
}

// ---------------------------------------------------------------------------
// Layer-1 pass 3: accum[d] += alpha * xl[s]. One wave per edge.
// ---------------------------------------------------------------------------
__global__ void edge_aggr1_kernel(const int* __restrict__ src,
                                  const int* __restrict__ dst,
                                  const float* __restrict__ logits,
                                  const float* __restrict__ lmax,
                                  const float* __restrict__ denom,
                                  const float* __restrict__ xl,
                                  float* __restrict__ accum,
                                  int E, int N) {
    int EN = E + N;
    int lane = threadIdx.x & 31;
    int eidx = blockIdx.x * (blockDim.x >> 5) + (threadIdx.x >> 5);
    if (eidx >= EN) return;
    int s, d;
    if (eidx < E) { s = src[eidx]; d = dst[eidx]; }
    else          { s = d = eidx - E; }
    float al[HEADS];
#pragma unroll
    for (int h = 0; h < HEADS; ++h) {
        float lg = logits[(long)eidx * HEADS + h];     // uniform -> broadcast
        float mx = lmax[(long)d * HEADS + h];
        float dn = denom[(long)d * HEADS + h];
        al[h] = expf(lg - mx) / (dn + 1e-16f);
    }
#pragma unroll
    for (int h = 0; h < HEADS; ++h) {
        int ch = h * CH + lane;
        atomicAdd(&accum[(long)d * FDIM + ch], al[h] * xl[(long)s * FDIM + ch]);
    }
}

__global__ void leaky_kernel(float* __restrict__ p, long n, float slope) {
    long i = (long)blockIdx.x * blockDim.x + threadIdx.x;
    if (i < n) p[i] = leaky(p[i], slope);
}

// ---------------------------------------------------------------------------
// Layer-2 edge pass 1 (C=1 per head): one thread per edge.
// ---------------------------------------------------------------------------
__global__ void edge_logits2_kernel(const int* __restrict__ src,
                                    const int* __restrict__ dst,
                                    const float* __restrict__ edge_attr,
                                    const float* __restrict__ loopA,
                                    const float* __restrict__ xl2,
                                    const float* __restrict__ xr2,
                                    const float* __restrict__ We2,
                                    const float* __restrict__ att2,
                                    float* __restrict__ logits,
                                    float* __restrict__ lmax,
                                    int E, int N) {
    __shared__ float sWe[EDIM * O2];
    __shared__ float sAtt[HEADS];
    for (int i = threadIdx.x; i < EDIM * O2; i += blockDim.x) sWe[i] = We2[i];
    if (threadIdx.x < HEADS) sAtt[threadIdx.x] = att2[threadIdx.x];
    __syncthreads();

    int EN = E + N;
    int idx = blockIdx.x * blockDim.x + threadIdx.x;
    if (idx >= EN) return;
    int s, d;
    const float* ea;
    if (idx < E) { s = src[idx]; d = dst[idx]; ea = edge_attr + (long)idx * EDIM; }
    else         { s = d = idx - E;            ea = loopA + (long)s * EDIM; }
    float ear[EDIM];
    const float4* ea4 = (const float4*)ea;
#pragma unroll
    for (int q = 0; q < EDIM / 4; ++q) {
        float4 v = ea4[q];
        ear[4 * q + 0] = v.x; ear[4 * q + 1] = v.y;
        ear[4 * q + 2] = v.z; ear[4 * q + 3] = v.w;
    }
#pragma unroll
    for (int h = 0; h < HEADS; ++h) {
        float v = xl2[(long)s * O2 + h] + xr2[(long)d * O2 + h];
#pragma unroll
        for (int k = 0; k < EDIM; ++k) v = fmaf(ear[k], sWe[k * O2 + h], v);
        float lg = leaky(v, SLOPE_ATT) * sAtt[h];
        logits[(long)idx * HEADS + h] = lg;
        atomic_max_f(&lmax[(long)d * HEADS + h], lg);
    }
}

__global__ void edge_aggr2_kernel(const int* __restrict__ src,
                                  const int* __restrict__ dst,
                                  const float* __restrict__ logits,
                                  const float* __restrict__ lmax,
                                  const float* __restrict__ denom,
                                  const float* __restrict__ xl2,
                                  float* __restrict__ accum,
                                  int E, int N) {
    int EN = E + N;
    int idx = blockIdx.x * blockDim.x + threadIdx.x;
    if (idx >= EN) return;
    int s, d;
    if (idx < E) { s = src[idx]; d = dst[idx]; }
    else         { s = d = idx - E; }
#pragma unroll
    for (int h = 0; h < HEADS; ++h) {
        float a = expf(logits[(long)idx * HEADS + h] - lmax[(long)d * HEADS + h]) /
                  (denom[(long)d * HEADS + h] + 1e-16f);
        atomicAdd(&accum[(long)d * O2 + h], a * xl2[(long)s * O2 + h]);
    }
}

__global__ void final_kernel(const float* __restrict__ accum,
                             float* __restrict__ out, int N) {
    int i = blockIdx.x * blockDim.x + threadIdx.x;
    if (i >= N) return;
    float s = 0.f;
#pragma unroll
    for (int h = 0; h < HEADS; ++h) s += accum[(long)i * O2 + h];
    out[i] = s * (1.f / HEADS);
}

// ---------------------------------------------------------------------------
// launch
// ---------------------------------------------------------------------------
extern "C" void kernel_launch(void* const* d_in, const int* in_sizes, int n_in,
                              void* d_out, int out_size, void* d_ws, size_t ws_size,
                              hipStream_t stream) {
    const float* x    = (const float*)d_in[0];
    const int*   ei   = (const int*)d_in[1];
    const float* eatt = (const float*)d_in[2];
    const float* Wl1  = (const float*)d_in[3];
    const float* Wr1  = (const float*)d_in[4];
    const float* We1  = (const float*)d_in[5];
    const float* att1 = (const float*)d_in[6];
    const float* Wl2  = (const float*)d_in[7];
    const float* Wr2  = (const float*)d_in[8];
    const float* We2  = (const float*)d_in[9];
    const float* att2 = (const float*)d_in[10];
    float* out = (float*)d_out;

    const int N  = in_sizes[0] / FDIM;
    const int E  = in_sizes[1] / 2;
    const int EN = E + N;
    const int* src = ei;
    const int* dst = ei + E;

    // workspace carve-out
    float* ws = (float*)d_ws;
    size_t o = 0;
    float* deg    = ws + o; o += (size_t)N;
    float* loopA  = ws + o; o += (size_t)N * EDIM;
    float* xl1    = ws + o; o += (size_t)N * FDIM;
    float* xr1    = ws + o; o += (size_t)N * FDIM;
    float* acc1   = ws + o; o += (size_t)N * FDIM;     // later: h (post-leaky)
    float* lmax1  = ws + o; o += (size_t)N * HEADS;
    float* den1   = ws + o; o += (size_t)N * HEADS;
    float* logitsS= ws + o; o += (size_t)EN * HEADS;   // shared across layers
    float* xl2    = ws + o; o += (size_t)N * O2;
    float* xr2    = ws + o; o += (size_t)N * O2;
    float* lmax2  = ws + o; o += (size_t)N * HEADS;
    float* den2   = ws + o; o += (size_t)N * HEADS;
    float* acc2   = ws + o; o += (size_t)N * O2;

    const int TB = 256;
    auto blk1 = [&](long n) { return (int)((n + TB - 1) / TB); };           // thread-per-elem
    auto blkW = [&](long n) { return (int)((n + (TB / 32) - 1) / (TB / 32)); }; // wave-per-elem

    // --- self-loop edge attr (mean of incoming) ---
    fill_kernel<<<blk1((long)N), TB, 0, stream>>>(deg, 0.f, N);
    fill_kernel<<<blk1((long)N * EDIM), TB, 0, stream>>>(loopA, 0.f, (long)N * EDIM);
    edge_prep_kernel<<<blk1(E), TB, 0, stream>>>(dst, eatt, deg, loopA, E);
    loop_norm_kernel<<<blk1(N), TB, 0, stream>>>(loopA, deg, N);

    // --- layer 1 GEMMs (WMMA f32 16x16x4) ---
    {
        int tiles = ((N + 15) / 16) * (FDIM / 16);
        int blocks = (tiles + 3) / 4;
        gemm_f32_wmma<<<blocks, 128, 0, stream>>>(x, Wl1, xl1, N, FDIM);
        gemm_f32_wmma<<<blocks, 128, 0, stream>>>(x, Wr1, xr1, N, FDIM);
    }

    // --- layer 1 segment softmax + aggregate ---
    fill_kernel<<<blk1((long)N * HEADS), TB, 0, stream>>>(lmax1, -3.402823466e38f, (long)N * HEADS);
    fill_kernel<<<blk1((long)N * HEADS), TB, 0, stream>>>(den1, 0.f, (long)N * HEADS);
    fill_kernel<<<blk1((long)N * FDIM), TB, 0, stream>>>(acc1, 0.f, (long)N * FDIM);
    edge_logits1_kernel<<<blkW(EN), TB, 0, stream>>>(src, dst, eatt, loopA, xl1, xr1,
                                                     We1, att1, logitsS, lmax1, E, N);
    edge_denom_kernel<<<blk1(EN), TB, 0, stream>>>(logitsS, lmax1, den1, dst, E, N);
    edge_aggr1_kernel<<<blkW(EN), TB, 0, stream>>>(src, dst, logitsS, lmax1, den1,
                                                   xl1, acc1, E, N);
    leaky_kernel<<<blk1((long)N * FDIM), TB, 0, stream>>>(acc1, (long)N * FDIM, SLOPE_ACT);

    // --- layer 2 GEMMs (Nout=4, padded tile) ---
    {
        int tiles = ((N + 15) / 16) * 1;
        int blocks = (tiles + 3) / 4;
        gemm_f32_wmma<<<blocks, 128, 0, stream>>>(acc1, Wl2, xl2, N, O2);
        gemm_f32_wmma<<<blocks, 128, 0, stream>>>(acc1, Wr2, xr2, N, O2);
    }

    // --- layer 2 segment softmax + aggregate + head-mean ---
    fill_kernel<<<blk1((long)N * HEADS), TB, 0, stream>>>(lmax2, -3.402823466e38f, (long)N * HEADS);
    fill_kernel<<<blk1((long)N * HEADS), TB, 0, stream>>>(den2, 0.f, (long)N * HEADS);
    fill_kernel<<<blk1((long)N * O2), TB, 0, stream>>>(acc2, 0.f, (long)N * O2);
    edge_logits2_kernel<<<blk1(EN), TB, 0, stream>>>(src, dst, eatt, loopA, xl2, xr2,
                                                     We2, att2, logitsS, lmax2, E, N);
    edge_denom_kernel<<<blk1(EN), TB, 0, stream>>>(logitsS, lmax2, den2, dst, E, N);
    edge_aggr2_kernel<<<blk1(EN), TB, 0, stream>>>(src, dst, logitsS, lmax2, den2,
                                                   xl2, acc2, E, N);
    final_kernel<<<blk1(N), TB, 0, stream>>>(acc2, out, N);

    (void)n_in; (void)out_size; (void)ws_size;
}